// HLRM_63376537420341
// MI455X (gfx1250) — compile-verified
//
#include <hip/hip_runtime.h>
#include <hip/hip_bf16.h>
#include <math.h>

// Problem constants (match reference)
#define B_  4096
#define M_  100
#define E_  128
#define R_  64

typedef float v2f __attribute__((ext_vector_type(2)));
typedef float v8f __attribute__((ext_vector_type(8)));

// ---------------- LDS layout (floats) ----------------
#define STRIDE    132            // 128 + 4 pad -> conflict-free column access
#define MT        112            // M padded to 7 WMMA tiles of 16
#define OFF_INTER 0
#define OFF_U     (OFF_INTER + MT * STRIDE)   // 14784
#define OFF_IP    (OFF_U + E_)
#define OFF_INN   (OFF_IP + E_)
#define OFF_W     (OFF_INN + E_)              // 4 weight vectors, 4*128
#define OFF_WSUA  (OFF_W + 4 * E_)
#define OFF_WSIA  (OFF_WSUA + E_)
#define OFF_LOGIT (OFF_WSIA + E_)             // 4 * 112 (logits, then attn in place)
#define OFF_SACC  (OFF_LOGIT + 4 * MT)        // 4 * 128
#define SMEM_FLOATS (OFF_SACC + 4 * E_)       // = 16896 floats = 67584 bytes

__device__ __forceinline__ float wave_sum(float v) {
    #pragma unroll
    for (int off = 16; off > 0; off >>= 1) v += __shfl_xor(v, off, 32);
    return v;
}
__device__ __forceinline__ float wave_max(float v) {
    #pragma unroll
    for (int off = 16; off > 0; off >>= 1) v = fmaxf(v, __shfl_xor(v, off, 32));
    return v;
}

// max-norm renormalized row load: 32 lanes x float4 = 128 floats
__device__ __forceinline__ void renorm_row(const float* __restrict__ src,
                                           float* __restrict__ dst, int lane) {
    float4 x = ((const float4*)src)[lane];
    float ss = x.x * x.x + x.y * x.y + x.z * x.z + x.w * x.w;
    ss = wave_sum(ss);
    float n = sqrtf(ss);
    float s = (n > 1.0f) ? (1.0f / (n + 1e-7f)) : 1.0f;
    x.x *= s; x.y *= s; x.z *= s; x.w *= s;
    ((float4*)dst)[lane] = x;
}

// -------- kernel 1: wsum_ua / wsum_ia = sum_r key[r,:]*val[r,:] --------
__global__ void hlrm_wsum_kernel(const float* __restrict__ ua_key,
                                 const float* __restrict__ ua_val,
                                 const float* __restrict__ ia_key,
                                 const float* __restrict__ ia_val,
                                 float* __restrict__ wsum) {
    int e = threadIdx.x;            // 128 threads
    float s1 = 0.f, s2 = 0.f;
    #pragma unroll 4
    for (int r = 0; r < R_; ++r) {
        s1 += ua_key[r * E_ + e] * ua_val[r * E_ + e];
        s2 += ia_key[r * E_ + e] * ia_val[r * E_ + e];
    }
    wsum[e]       = s1;
    wsum[E_ + e]  = s2;
}

// -------- kernel 2: one workgroup (8 wave32) per batch element --------
__global__ void __launch_bounds__(256, 2)
hlrm_main_kernel(const int* __restrict__ user_id,
                 const int* __restrict__ item_id_p,
                 const int* __restrict__ item_id_n,
                 const int* __restrict__ inter_id,
                 const float* __restrict__ user_emb_w,
                 const float* __restrict__ item_emb_w,
                 const float* __restrict__ wsum_g,
                 float* __restrict__ out) {
    extern __shared__ float sm[];
    float* sm_inter = sm + OFF_INTER;
    float* sm_u     = sm + OFF_U;
    float* sm_ip    = sm + OFF_IP;
    float* sm_inn   = sm + OFF_INN;
    float* sm_w     = sm + OFF_W;
    float* sm_wsua  = sm + OFF_WSUA;
    float* sm_wsia  = sm + OFF_WSIA;
    float* sm_logit = sm + OFF_LOGIT;
    float* sm_sacc  = sm + OFF_SACC;

    const int b    = blockIdx.x;
    const int t    = threadIdx.x;
    const int wave = t >> 5;
    const int lane = t & 31;
    const int half = lane >> 4;
    const int lm   = lane & 15;

    // ---- Stage 1: renorm lookups of u / ip / inn (waves 0..2) ----
    if (wave == 0)      renorm_row(user_emb_w + (size_t)user_id[b]   * E_, sm_u,   lane);
    else if (wave == 1) renorm_row(item_emb_w + (size_t)item_id_p[b] * E_, sm_ip,  lane);
    else if (wave == 2) renorm_row(item_emb_w + (size_t)item_id_n[b] * E_, sm_inn, lane);

    // ---- Stage 2: gather + renorm inter rows into LDS (all 8 waves) ----
    for (int m = wave; m < MT; m += 8) {
        if (m < M_) {
            const float* src = item_emb_w + (size_t)inter_id[b * M_ + m] * E_;
            if (m + 8 < M_) {
                const float* nxt = item_emb_w + (size_t)inter_id[b * M_ + m + 8] * E_;
                __builtin_prefetch(nxt + 4 * lane, 0, 3);   // global_prefetch_b8
            }
            renorm_row(src, sm_inter + m * STRIDE, lane);
        } else {
            float4 z = {0.f, 0.f, 0.f, 0.f};
            ((float4*)(sm_inter + m * STRIDE))[lane] = z;   // zero pad rows 100..111
        }
    }
    __syncthreads();

    // ---- Stage 3: weight vectors w_k[e] and wsum copies ----
    if (t < E_) {
        const int e = t;
        const float uu = sm_u[e], pp = sm_ip[e], nn = sm_inn[e];
        const float wa = wsum_g[e], wi = wsum_g[E_ + e];
        sm_wsua[e] = wa;
        sm_wsia[e] = wi;
        sm_w[0 * E_ + e] = uu * pp * wa;   // (ua, p): mul=ip, attn=u
        sm_w[1 * E_ + e] = uu * pp * wi;   // (ia, p): mul=u,  attn=ip
        sm_w[2 * E_ + e] = uu * nn * wa;   // (ua, n): mul=inn,attn=u
        sm_w[3 * E_ + e] = uu * nn * wi;   // (ia, n): mul=u,  attn=inn
    }
    __syncthreads();

    // ---- Stage 4: logits[m,k] = sum_e inter[m,e]*w_k[e] via f32 WMMA ----
    // C tile: rows M = inter-row m (16/tile), cols N = weight index k (0..3 live).
    if (wave < 7) {
        const int   tile = wave;
        const float* Arow = sm_inter + (tile * 16 + lm) * STRIDE;
        const float  bmask = (lm < 4) ? 1.0f : 0.0f;
        const float* wrow  = sm_w + (lm < 4 ? lm : 0) * E_;
        v8f c = {0.f, 0.f, 0.f, 0.f, 0.f, 0.f, 0.f, 0.f};
        #pragma unroll 8
        for (int k0 = 0; k0 < E_; k0 += 4) {
            const int ke = k0 + 2 * half;     // A 16x4 f32 layout: halves hold K pairs
            v2f a, bb;
            a[0]  = Arow[ke];           a[1]  = Arow[ke + 1];
            bb[0] = wrow[ke] * bmask;   bb[1] = wrow[ke + 1] * bmask;
            c = __builtin_amdgcn_wmma_f32_16x16x4_f32(false, a, false, bb,
                                                      (short)0, c, false, false);
        }
        if (lm < 4) {                          // N = lm = weight index k
            #pragma unroll
            for (int v = 0; v < 8; ++v)        // M = v + 8*half
                sm_logit[lm * MT + tile * 16 + v + 8 * half] = c[v];
        }
    }
    __syncthreads();

    // ---- Stage 5: softmax over m (4 waves, one per logit vector) ----
    if (wave < 4) {
        float* L = sm_logit + wave * MT;
        float vals[4], mx = -INFINITY;
        #pragma unroll
        for (int i = 0; i < 4; ++i) {
            int m = lane + 32 * i;
            vals[i] = (m < M_) ? L[m] : -INFINITY;
            mx = fmaxf(mx, vals[i]);
        }
        mx = wave_max(mx);
        float se = 0.f;
        #pragma unroll
        for (int i = 0; i < 4; ++i) {
            int m = lane + 32 * i;
            vals[i] = (m < M_) ? expf(vals[i] - mx) : 0.f;
            se += vals[i];
        }
        se = wave_sum(se);
        const float inv = 1.0f / se;
        #pragma unroll
        for (int i = 0; i < 4; ++i) {
            int m = lane + 32 * i;
            if (m < MT) L[m] = vals[i] * inv;  // attn in place; pad rows -> 0
        }
    }
    __syncthreads();

    // ---- Stage 6: sacc_k[e] = sum_m attn_k[m]*inter[m,e] via f32 WMMA ----
    // C tile: rows M = k (4 live), cols N = e-local (16/tile, 8 tiles = 8 waves).
    {
        const int   etile = wave;
        const float amask = (lm < 4) ? 1.0f : 0.0f;
        const float* Lrow = sm_logit + (lm < 4 ? lm : 0) * MT;
        v8f c = {0.f, 0.f, 0.f, 0.f, 0.f, 0.f, 0.f, 0.f};
        #pragma unroll 7
        for (int m0 = 0; m0 < MT; m0 += 4) {
            const int mk = m0 + 2 * half;
            v2f a, bb;
            a[0]  = Lrow[mk] * amask;     a[1]  = Lrow[mk + 1] * amask;
            bb[0] = sm_inter[mk * STRIDE + etile * 16 + lm];
            bb[1] = sm_inter[(mk + 1) * STRIDE + etile * 16 + lm];
            c = __builtin_amdgcn_wmma_f32_16x16x4_f32(false, a, false, bb,
                                                      (short)0, c, false, false);
        }
        if (half == 0) {                       // rows M = v (only v<4 live)
            #pragma unroll
            for (int v = 0; v < 4; ++v)
                sm_sacc[v * E_ + etile * 16 + lm] = c[v];
        }
    }
    __syncthreads();

    // ---- Stage 7: combine + write all 5 outputs ----
    if (t < E_) {
        const int e = t;
        const float uu = sm_u[e], pp = sm_ip[e], nn = sm_inn[e];
        const float wa = sm_wsua[e], wi = sm_wsia[e];
        const float relp = pp * wa * sm_sacc[0 * E_ + e] + uu * wi * sm_sacc[1 * E_ + e];
        const float reln = nn * wa * sm_sacc[2 * E_ + e] + uu * wi * sm_sacc[3 * E_ + e];
        const size_t BE = (size_t)B_ * E_;
        const size_t base = (size_t)b * E_ + e;
        out[0 * BE + base] = uu;
        out[1 * BE + base] = pp;
        out[2 * BE + base] = nn;
        out[3 * BE + base] = relp;
        out[4 * BE + base] = reln;
    }
}

extern "C" void kernel_launch(void* const* d_in, const int* in_sizes, int n_in,
                              void* d_out, int out_size, void* d_ws, size_t ws_size,
                              hipStream_t stream) {
    const int*   user_id    = (const int*)d_in[0];
    const int*   item_id_p  = (const int*)d_in[1];
    const int*   item_id_n  = (const int*)d_in[2];
    const int*   inter_id   = (const int*)d_in[3];
    const float* user_emb_w = (const float*)d_in[4];
    const float* item_emb_w = (const float*)d_in[5];
    const float* ua_key     = (const float*)d_in[6];
    const float* ua_val     = (const float*)d_in[7];
    const float* ia_key     = (const float*)d_in[8];
    const float* ia_val     = (const float*)d_in[9];
    float* out  = (float*)d_out;
    float* wsum = (float*)d_ws;            // 2*E floats of scratch

    hlrm_wsum_kernel<<<1, E_, 0, stream>>>(ua_key, ua_val, ia_key, ia_val, wsum);
    hlrm_main_kernel<<<B_, 256, SMEM_FLOATS * sizeof(float), stream>>>(
        user_id, item_id_p, item_id_n, inter_id,
        user_emb_w, item_emb_w, wsum, out);
}